// Ipfl_15839839388070
// MI455X (gfx1250) — compile-verified
//
#include <hip/hip_runtime.h>
#include <math.h>

typedef __attribute__((ext_vector_type(2))) float v2f;
typedef __attribute__((ext_vector_type(8))) float v8f;

#define N_TOT    4096
#define D_DIM    256
#define SCALE_F  10.0f
#define EPS_F    1e-5f
#define MARG_I   40.0f
#define MARG_J   6.0f
#define K_I      4
#define K_J      6
#define BIGF     1e30f

#define SA       258            // As row stride (floats), padded for LDS banks
#define SB       66             // Bs row stride (floats), K-chunk=64 + pad
#define PANEL    128            // columns per panel (one 16-col tile per wave)
#define KCH      64             // K chunk staged in LDS

// shared-memory layout (float offsets)
#define OFF_AS     0                         // 16*258      = 4128
#define OFF_BS     (OFF_AS + 16*SA)          // 128*66      = 8448
#define OFF_UNION  (OFF_BS + PANEL*SB)       // max(dist, merge) = 3616
#define OFF_DIST   OFF_UNION                 // 8 waves * 16*17 = 2176 (aliases merge)
#define OFF_MIL    OFF_UNION                 // 16*16*4 = 1024
#define OFF_MJL    (OFF_MIL + 1024)          // 16*16*6 = 1536
#define OFF_MIMS   (OFF_MJL + 1536)          // 16*16*2 = 512
#define OFF_MJMS   (OFF_MIMS + 512)          // 16*16*2 = 512
#define OFF_RLOSS  (OFF_MJMS + 512)          // 16*2    = 32
#define OFF_SQROW  (OFF_UNION + 3616)        // 16
#define OFF_CAMROW (OFF_SQROW + 16)          // 16
#define SMEM_FLOATS (OFF_CAMROW + 16)        // 16224 floats = 64896 B (< 64 KB)

// Streaming "k smallest + logsumexp of the rest".
// Branchless insert: x propagates through the sorted-ascending list via
// min/max; the evicted max (either d or old lst[K-1]) feeds the online LSE.
template <int K>
__device__ __forceinline__ void stream_update(float (&lst)[K], float& m, float& s, float d) {
    float x = d;
#pragma unroll
    for (int j = 0; j < K; ++j) {
        float lo = fminf(lst[j], x);
        x = fmaxf(lst[j], x);
        lst[j] = lo;
    }
    if (x < BIGF) {                 // skip +inf sentinels (empty slots)
        float e = -SCALE_F * x;
        if (e > m) { s = s * expf(m - e) + 1.0f; m = e; }
        else       { s += expf(e - m); }
    }
}

// -------- kernel 1: row squared norms --------
__global__ __launch_bounds__(256) void sqnorm_kernel(const float* __restrict__ feat,
                                                     float* __restrict__ sqn) {
    __shared__ float red[256];
    const int tid = threadIdx.x;
    const int row = blockIdx.x * 16 + (tid >> 4);
    const int seg = tid & 15;
    const float4* f4 = (const float4*)(feat + row * D_DIM + seg * 16);
    float acc = 0.0f;
#pragma unroll
    for (int i = 0; i < 4; ++i) {
        float4 v = f4[i];
        acc += v.x * v.x + v.y * v.y + v.z * v.z + v.w * v.w;
    }
    red[tid] = acc;
    __syncthreads();
    if (seg == 0) {
        float s = 0.0f;
#pragma unroll
        for (int j = 0; j < 16; ++j) s += red[tid + j];
        sqn[row] = s;
    }
}

// -------- kernel 2: fused WMMA gram -> dist -> streaming top-k/LSE --------
__global__ __launch_bounds__(256) void ipfl_main(const float* __restrict__ feat,
                                                 const int* __restrict__ camid,
                                                 const float* __restrict__ sqn,
                                                 float* __restrict__ partials) {
    __shared__ float smem[SMEM_FLOATS];
    const int tid = threadIdx.x;
    const int w  = tid >> 5;      // wave id 0..7
    const int l  = tid & 31;      // lane
    const int lh = l >> 4;        // lane half (0/1)
    const int ln = l & 15;
    const int rowBase = blockIdx.x * 16;

    // stage A tile: 16 rows x 256 K (1024 float4 chunks)
#pragma unroll
    for (int i = 0; i < 4; ++i) {
        int v = tid + i * 256;
        int r = v >> 6, q = v & 63;
        float4 f = ((const float4*)(feat + (size_t)(rowBase + r) * D_DIM))[q];
        float* dst = &smem[OFF_AS + r * SA + q * 4];
        dst[0] = f.x; dst[1] = f.y; dst[2] = f.z; dst[3] = f.w;
    }
    if (tid < 16) {
        smem[OFF_SQROW + tid] = sqn[rowBase + tid];
        ((int*)&smem[OFF_CAMROW])[tid] = camid[rowBase + tid];
    }

    // per-lane streaming state: row = ln, column subset = (panel,w,lh)
    float listI[K_I], listJ[K_J];
#pragma unroll
    for (int j = 0; j < K_I; ++j) listI[j] = BIGF;
#pragma unroll
    for (int j = 0; j < K_J; ++j) listJ[j] = BIGF;
    float mI = -BIGF, sI = 0.0f, mJ = -BIGF, sJ = 0.0f;

    for (int panel = 0; panel < N_TOT / PANEL; ++panel) {
        const int colPanel = panel * PANEL;
        if (panel + 1 < N_TOT / PANEL)   // warm next panel in L2 (global_prefetch_b8)
            __builtin_prefetch(feat + (size_t)(colPanel + PANEL) * D_DIM + tid * 64, 0, 1);

        v8f acc = {};
        for (int kc = 0; kc < D_DIM / KCH; ++kc) {
            __syncthreads();     // previous chunk fully consumed
            // stage B chunk: 128 cols x 64 K (2048 float4 chunks)
#pragma unroll
            for (int i = 0; i < 8; ++i) {
                int v = tid + i * 256;
                int c = v >> 4, q = v & 15;
                float4 f = ((const float4*)(feat + (size_t)(colPanel + c) * D_DIM + kc * KCH))[q];
                float* dst = &smem[OFF_BS + c * SB + q * 4];
                dst[0] = f.x; dst[1] = f.y; dst[2] = f.z; dst[3] = f.w;
            }
            __syncthreads();
            const float* Ab = &smem[OFF_AS + ln * SA + kc * KCH + lh * 2];
            const float* Bb = &smem[OFF_BS + (w * 16 + ln) * SB + lh * 2];
#pragma unroll
            for (int t = 0; t < 16; ++t) {
                v2f a = *(const v2f*)(Ab + t * 4);
                v2f b = *(const v2f*)(Bb + t * 4);
                acc = __builtin_amdgcn_wmma_f32_16x16x4_f32(
                    false, a, false, b, (short)0, acc, false, false);
            }
        }

        // gram -> dist, encode cam mask in sign bit, bounce through LDS
        const int colg = colPanel + w * 16 + ln;      // N index for this lane
        const float sqc = sqn[colg];
        const int camc = camid[colg];
        float* distW = &smem[OFF_DIST + w * (16 * 17)];
#pragma unroll
        for (int r = 0; r < 8; ++r) {
            const int m = r + 8 * lh;                  // C layout: VGPR r, half lh
            float d2 = smem[OFF_SQROW + m] + sqc + EPS_F - 2.0f * acc[r];
            float d = sqrtf(fmaxf(d2, 1e-12f));
            bool same = (((const int*)&smem[OFF_CAMROW])[m] == camc);
            distW[m * 17 + ln] = same ? d : -d;        // d >= 1e-6 > 0
        }
        __syncthreads();
        // transpose consume: this lane owns row ln, cols lh*8..lh*8+7 of the tile
#pragma unroll
        for (int j = 0; j < 8; ++j) {
            float v = distW[ln * 17 + lh * 8 + j];
            if (v > 0.0f) stream_update<K_I>(listI, mI, sI, v);
            else          stream_update<K_J>(listJ, mJ, sJ, -v);
        }
    }

    // -------- merge 16 partial states per row --------
    __syncthreads();
    {
        const int contrib = w * 2 + lh;
        const int rowm = ln;
#pragma unroll
        for (int j = 0; j < K_I; ++j) smem[OFF_MIL + rowm * 64 + contrib * 4 + j] = listI[j];
#pragma unroll
        for (int j = 0; j < K_J; ++j) smem[OFF_MJL + rowm * 96 + contrib * 6 + j] = listJ[j];
        smem[OFF_MIMS + rowm * 32 + contrib * 2 + 0] = mI;
        smem[OFF_MIMS + rowm * 32 + contrib * 2 + 1] = sI;
        smem[OFF_MJMS + rowm * 32 + contrib * 2 + 0] = mJ;
        smem[OFF_MJMS + rowm * 32 + contrib * 2 + 1] = sJ;
    }
    __syncthreads();

    if (tid < 16) {
        const int row = tid;
        // intra
        float L[K_I];
#pragma unroll
        for (int j = 0; j < K_I; ++j) L[j] = BIGF;
        float M = -BIGF, S = 0.0f;
        for (int c = 0; c < 16; ++c)
            for (int j = 0; j < K_I; ++j)
                stream_update<K_I>(L, M, S, smem[OFF_MIL + row * 64 + c * 4 + j]);
        for (int c = 0; c < 16; ++c) {
            float mc = smem[OFF_MIMS + row * 32 + c * 2];
            float sc = smem[OFF_MIMS + row * 32 + c * 2 + 1];
            if (sc > 0.0f) {
                if (mc > M) { S = S * expf(M - mc) + sc; M = mc; }
                else        { S += sc * expf(mc - M); }
            }
        }
        float lossI = 0.0f;
        if (S > 0.0f && L[0] < BIGF)
            lossI = fmaxf(SCALE_F * L[0] + (M + logf(S)) + MARG_I, 0.0f);

        // inter
        float Lj[K_J];
#pragma unroll
        for (int j = 0; j < K_J; ++j) Lj[j] = BIGF;
        float Mj = -BIGF, Sj = 0.0f;
        for (int c = 0; c < 16; ++c)
            for (int j = 0; j < K_J; ++j)
                stream_update<K_J>(Lj, Mj, Sj, smem[OFF_MJL + row * 96 + c * 6 + j]);
        for (int c = 0; c < 16; ++c) {
            float mc = smem[OFF_MJMS + row * 32 + c * 2];
            float sc = smem[OFF_MJMS + row * 32 + c * 2 + 1];
            if (sc > 0.0f) {
                if (mc > Mj) { Sj = Sj * expf(Mj - mc) + sc; Mj = mc; }
                else         { Sj += sc * expf(mc - Mj); }
            }
        }
        float lossJ = 0.0f;
        if (Sj > 0.0f && Lj[0] < BIGF)
            lossJ = fmaxf(SCALE_F * Lj[0] + (Mj + logf(Sj)) + MARG_J, 0.0f);

        smem[OFF_RLOSS + row * 2 + 0] = lossI;
        smem[OFF_RLOSS + row * 2 + 1] = lossJ;
    }
    __syncthreads();
    if (tid == 0) {
        float a = 0.0f, b = 0.0f;
        for (int r = 0; r < 16; ++r) {
            a += smem[OFF_RLOSS + r * 2 + 0];
            b += smem[OFF_RLOSS + r * 2 + 1];
        }
        partials[blockIdx.x * 2 + 0] = a;
        partials[blockIdx.x * 2 + 1] = b;
    }
}

// -------- kernel 3: deterministic final reduction --------
__global__ __launch_bounds__(256) void finalize_kernel(const float* __restrict__ partials,
                                                       float* __restrict__ out) {
    __shared__ float sI[256], sJ[256];
    const int t = threadIdx.x;
    sI[t] = partials[t * 2 + 0];
    sJ[t] = partials[t * 2 + 1];
    __syncthreads();
    for (int o = 128; o > 0; o >>= 1) {
        if (t < o) { sI[t] += sI[t + o]; sJ[t] += sJ[t + o]; }
        __syncthreads();
    }
    if (t == 0) out[0] = (sI[0] + 0.5f * sJ[0]) / (float)N_TOT;
}

extern "C" void kernel_launch(void* const* d_in, const int* in_sizes, int n_in,
                              void* d_out, int out_size, void* d_ws, size_t ws_size,
                              hipStream_t stream) {
    (void)in_sizes; (void)n_in; (void)out_size; (void)ws_size;
    const float* feat  = (const float*)d_in[0];
    const int*   camid = (const int*)d_in[1];
    float* sqn      = (float*)d_ws;           // 4096 floats
    float* partials = sqn + N_TOT;            // 256*2 floats
    float* out      = (float*)d_out;

    sqnorm_kernel<<<N_TOT / 16, 256, 0, stream>>>(feat, sqn);
    ipfl_main<<<N_TOT / 16, 256, 0, stream>>>(feat, camid, sqn, partials);
    finalize_kernel<<<1, 256, 0, stream>>>(partials, out);
}